// RegionCrossAttention_61263413510176
// MI455X (gfx1250) — compile-verified
//
#include <hip/hip_runtime.h>
#include <hip/hip_bf16.h>
#include <stdint.h>

// Problem constants (match reference)
#define B_  8
#define C_  256
#define K_  8
#define H_  128
#define W_  128
#define HW_ (H_ * W_)

typedef __attribute__((ext_vector_type(2))) float v2f;
typedef __attribute__((ext_vector_type(8))) float v8f;

// ---------------------------------------------------------------------------
// Kernel 1: area[b][k] = sum over H*W of mask[b][k][:]
// grid = B*K blocks, 256 threads
// ---------------------------------------------------------------------------
__global__ __launch_bounds__(256) void area_kernel(const float* __restrict__ mask,
                                                   float* __restrict__ area) {
  __shared__ float red[256];
  const int bk = blockIdx.x;
  const float* m = mask + (size_t)bk * HW_;
  float s = 0.f;
  for (int i = threadIdx.x; i < HW_; i += 256) s += m[i];
  red[threadIdx.x] = s;
  __syncthreads();
  for (int off = 128; off > 0; off >>= 1) {
    if (threadIdx.x < off) red[threadIdx.x] += red[threadIdx.x + off];
    __syncthreads();
  }
  if (threadIdx.x == 0) area[bk] = red[0];
}

// ---------------------------------------------------------------------------
// Kernel 2: fold the weight chain into small per-(b,k) vectors:
//   g      = sqrt(p / max_k p), p = area/(H*W)
//   vtok   = region_vec * g
//   k_tok  = k_w @ vtok + k_b ;  v_tok = v_w @ vtok + v_b
//   w2[c]  = (sum_o q_w[o][c] * k_tok[o]) / 16        (16 = sqrt(C))
//   b2     = (sum_o q_b[o]    * k_tok[o]) / 16
//   pv[c]  =  sum_o proj_w[c][o] * v_tok[o]
// grid = B*K blocks, 256 threads (thread = channel)
// ---------------------------------------------------------------------------
__global__ __launch_bounds__(256) void fold_kernel(
    const float* __restrict__ area, const float* __restrict__ region_vec,
    const float* __restrict__ q_w, const float* __restrict__ q_b,
    const float* __restrict__ k_w, const float* __restrict__ k_b,
    const float* __restrict__ v_w, const float* __restrict__ v_b,
    const float* __restrict__ proj_w,
    float* __restrict__ w2, float* __restrict__ b2, float* __restrict__ pv) {
  __shared__ float vtok[C_], ktok[C_], vtok2[C_];
  const int b = blockIdx.x / K_;
  const int k = blockIdx.x % K_;
  const int t = threadIdx.x;

  const float inv_hw = 1.0f / (float)(HW_);
  float p = area[b * K_ + k] * inv_hw;
  float dmax = 1e-6f;
  for (int kk = 0; kk < K_; ++kk) dmax = fmaxf(dmax, area[b * K_ + kk] * inv_hw);
  float g = sqrtf(fmaxf(p / dmax, 0.0f));

  vtok[t] = region_vec[((size_t)b * K_ + k) * C_ + t] * g;
  __syncthreads();

  float kt = k_b[t], vt = v_b[t];
  for (int c = 0; c < C_; ++c) {
    kt += k_w[t * C_ + c] * vtok[c];
    vt += v_w[t * C_ + c] * vtok[c];
  }
  ktok[t] = kt;
  vtok2[t] = vt;
  __syncthreads();

  float w2v = 0.f, pvv = 0.f;
  for (int o = 0; o < C_; ++o) {
    w2v += q_w[o * C_ + t] * ktok[o];
    pvv += proj_w[t * C_ + o] * vtok2[o];
  }
  w2[((size_t)b * K_ + k) * C_ + t] = w2v * (1.0f / 16.0f);
  pv[((size_t)b * K_ + k) * C_ + t] = pvv;
  if (t == 0) {
    float s = 0.f;
    for (int o = 0; o < C_; ++o) s += q_b[o] * ktok[o];
    b2[b * K_ + k] = s * (1.0f / 16.0f);
  }
}

// ---------------------------------------------------------------------------
// Kernel 3: streaming attention kernel (memory-bound, WMMA + async-LDS)
//   Per wave: 16-pixel tile, feat tile (256ch x 16pix = 16KB) staged into
//   wave-private LDS with global_load_async_to_lds_b128 (32 instrs/tile).
//   Stage 1 : logits^T(16k x 16pix) = w2_pad(16x256) @ feat(256x16), 64 WMMAs
//   Softmax : lane-local over 8 VGPRs (lanes 0-15 hold valid pixels)
//   Stage 2 : delta^T(16ch x 16pix) = pv^T @ attn^T, 2 WMMAs per channel tile
//   Epilogue: out = feat + sigmoid(alpha)*(delta + proj_b), feat from LDS
// grid = (HW/32, B), 64 threads (2 waves)
// ---------------------------------------------------------------------------
__global__ __launch_bounds__(64) void attn_kernel(
    const float* __restrict__ feat, const float* __restrict__ mask,
    const float* __restrict__ w2g, const float* __restrict__ b2g,
    const float* __restrict__ pvg, const float* __restrict__ proj_b,
    const float* __restrict__ mbs_p, const float* __restrict__ alpha_p,
    float* __restrict__ out) {
  __shared__ float w2_s[16 * C_];                       // 16KB, rows 8..15 zero
  __shared__ float pv_s[K_ * C_];                       // 8KB
  __shared__ float pb_s[C_];                            // 1KB
  __shared__ float b2_s[16];
  __shared__ __align__(16) float feat_s[2 * C_ * 16];   // 32KB: per-wave tile

  const int b = blockIdx.y;
  const int tid = threadIdx.x;

  for (int i = tid; i < 16 * C_; i += 64) {
    int row = i >> 8;
    w2_s[i] = (row < K_) ? w2g[((size_t)b * K_ + row) * C_ + (i & (C_ - 1))] : 0.0f;
  }
  for (int i = tid; i < K_ * C_; i += 64) pv_s[i] = pvg[(size_t)b * K_ * C_ + i];
  for (int i = tid; i < C_; i += 64) pb_s[i] = proj_b[i];
  if (tid < 16) b2_s[tid] = (tid < K_) ? b2g[b * K_ + tid] : 0.0f;
  __syncthreads();

  const float mbs = mbs_p[0];
  const float gate = 1.0f / (1.0f + __expf(-alpha_p[0]));

  const int wave = tid >> 5;
  const int lane = tid & 31;
  const int half = lane >> 4;   // 0: lanes 0-15, 1: lanes 16-31
  const int row  = lane & 15;
  const int p0 = (blockIdx.x * 2 + wave) * 16;
  const int p  = p0 + row;                         // this lane's pixel column

  const float* fb = feat + (size_t)b * C_ * HW_;   // batch base

  // ---- Async-stage feat tile into wave-private LDS ------------------------
  // Each lane: 16B pixel-quad of one channel -> 8 channels per instruction.
  float* fsw = feat_s + wave * (C_ * 16);
  const uint32_t lds_base = (uint32_t)(uintptr_t)fsw;
  const int chl = lane >> 2;                       // 0..7: channel within block
  const int pq  = (lane & 3) * 4;                  // pixel quad
  const float* gsrc = fb + (size_t)chl * HW_ + p0 + pq;
  #pragma unroll 4
  for (int cb = 0; cb < C_; cb += 8) {
    uint32_t laddr = lds_base + (uint32_t)(((cb + chl) * 16 + pq) * 4);
    const float* ga = gsrc + (size_t)cb * HW_;
    asm volatile("global_load_async_to_lds_b128 %0, %1, off"
                 :: "v"(laddr), "v"(ga) : "memory");
  }
  asm volatile("s_wait_asynccnt 0" ::: "memory");

  // ---- Stage 1: logits^T = w2_pad @ feat_tile -----------------------------
  // A (16x4): M=k row (lane&15), VGPR0=K chans {c+2h}, VGPR1={c+2h+1}
  // B (4x16): K chans in lane-halves, N=pixel (from LDS)
  v8f acc = {};
  #pragma unroll 4
  for (int c = 0; c < C_; c += 4) {
    const int cc = c + 2 * half;
    v2f Aop, Bop;
    Aop.x = w2_s[row * C_ + cc];
    Aop.y = w2_s[row * C_ + cc + 1];
    Bop.x = fsw[cc * 16 + row];
    Bop.y = fsw[(cc + 1) * 16 + row];
    acc = __builtin_amdgcn_wmma_f32_16x16x4_f32(false, Aop, false, Bop,
                                                (short)0, acc, false, false);
  }

  // ---- Bias + softmax (lane-local; lanes 0-15 carry the real pixels) ------
  const float* mB = mask + (size_t)b * K_ * HW_ + p;
  float logit[8], attnv[8];
  float mx = -1e30f;
  #pragma unroll
  for (int r = 0; r < 8; ++r) {
    float l = acc[r] + b2_s[r] + mbs * __logf(mB[(size_t)r * HW_] + 1e-6f);
    logit[r] = l;
    mx = fmaxf(mx, l);
  }
  float sum = 0.f;
  #pragma unroll
  for (int r = 0; r < 8; ++r) {
    float e = __expf(logit[r] - mx);
    attnv[r] = e;
    sum += e;
  }
  const float inv = 1.0f / sum;
  #pragma unroll
  for (int r = 0; r < 8; ++r) attnv[r] *= inv;

  // ---- Build stage-2 B operands: attn^T (k x pixel) -----------------------
  // lanes 16-31 need attn[k=2,3,6,7] of lane (lane-16): shuffle + select.
  float t2 = __shfl(attnv[2], row, 32);
  float t3 = __shfl(attnv[3], row, 32);
  float t6 = __shfl(attnv[6], row, 32);
  float t7 = __shfl(attnv[7], row, 32);
  v2f B1, B2;
  B1.x = half ? t2 : attnv[0];
  B1.y = half ? t3 : attnv[1];
  B2.x = half ? t6 : attnv[4];
  B2.y = half ? t7 : attnv[5];

  // ---- Stage 2 + epilogue -------------------------------------------------
  // A (16x4): M=channel (ct+row), K = region index (2 WMMAs cover k=0..7)
  float* ob = out + (size_t)b * C_ * HW_;
  for (int ct = 0; ct < C_; ct += 16) {
    const int ch = ct + row;
    v2f A1, A2;
    A1.x = pv_s[(0 + 2 * half) * C_ + ch];
    A1.y = pv_s[(1 + 2 * half) * C_ + ch];
    A2.x = pv_s[(4 + 2 * half) * C_ + ch];
    A2.y = pv_s[(5 + 2 * half) * C_ + ch];
    v8f d = {};
    d = __builtin_amdgcn_wmma_f32_16x16x4_f32(false, A1, false, B1,
                                              (short)0, d, false, false);
    d = __builtin_amdgcn_wmma_f32_16x16x4_f32(false, A2, false, B2,
                                              (short)0, d, false, false);
    #pragma unroll
    for (int r = 0; r < 8; ++r) {
      const int c_out = ct + r + 8 * half;            // D row -> channel
      const float f = fsw[c_out * 16 + row];          // feat from LDS tile
      ob[(size_t)c_out * HW_ + p] = f + gate * (d[r] + pb_s[c_out]);
    }
  }
}

// ---------------------------------------------------------------------------
extern "C" void kernel_launch(void* const* d_in, const int* in_sizes, int n_in,
                              void* d_out, int out_size, void* d_ws, size_t ws_size,
                              hipStream_t stream) {
  const float* feat       = (const float*)d_in[0];
  const float* mask       = (const float*)d_in[1];
  const float* region_vec = (const float*)d_in[2];
  const float* q_w        = (const float*)d_in[3];
  const float* q_b        = (const float*)d_in[4];
  const float* k_w        = (const float*)d_in[5];
  const float* k_b        = (const float*)d_in[6];
  const float* v_w        = (const float*)d_in[7];
  const float* v_b        = (const float*)d_in[8];
  const float* proj_w     = (const float*)d_in[9];
  const float* proj_b     = (const float*)d_in[10];
  const float* mbs        = (const float*)d_in[11];
  const float* alpha      = (const float*)d_in[12];
  float* out = (float*)d_out;

  float* ws   = (float*)d_ws;
  float* area = ws;                       // B*K       = 64
  float* b2   = ws + 64;                  // B*K       = 64
  float* w2   = ws + 128;                 // B*K*C     = 16384
  float* pv   = ws + 128 + B_ * K_ * C_;  // B*K*C     = 16384

  area_kernel<<<B_ * K_, 256, 0, stream>>>(mask, area);
  fold_kernel<<<B_ * K_, 256, 0, stream>>>(area, region_vec, q_w, q_b, k_w, k_b,
                                           v_w, v_b, proj_w, w2, b2, pv);
  dim3 grid(HW_ / 32, B_);
  attn_kernel<<<grid, 64, 0, stream>>>(feat, mask, w2, b2, pv, proj_b,
                                       mbs, alpha, out);
}